// TestModuleV3_22874995818881
// MI455X (gfx1250) — compile-verified
//
#include <hip/hip_runtime.h>
#include <hip/hip_bf16.h>

typedef __attribute__((ext_vector_type(2))) float v2f;
typedef __attribute__((ext_vector_type(4))) float v4f;
typedef __attribute__((ext_vector_type(8))) float v8f;

#define V_VOCAB 50257
#define D_DIM   512
#define S_DIM   8192
#define A_DIM   256
#define T_CTX   4096
#define MAXSYN  8
#define KX      (D_DIM + A_DIM)   // 768

#define WMMA_F32(a, b, c) __builtin_amdgcn_wmma_f32_16x16x4_f32( \
    false, (a), false, (b), (short)0, (c), false, false)

__device__ __forceinline__ float sigmoidf_(float z) {
    return 1.0f / (1.0f + __expf(-z));
}

// ---------------------------------------------------------------------------
// g = sigmoid(emb_v2[ids] - 4) - sigmoid(-emb_v2[ids] - 4)   -> [T, D]
// ---------------------------------------------------------------------------
__global__ void k_gather_g(const int* __restrict__ ids,
                           const float* __restrict__ emb_v2,
                           float* __restrict__ g_out) {
    int i = blockIdx.x * blockDim.x + threadIdx.x;     // exactly T*D threads
    int t = i >> 9;                                    // / 512
    int d = i & (D_DIM - 1);
    float x = emb_v2[(size_t)ids[t] * D_DIM + d];
    g_out[i] = sigmoidf_(x - 4.0f) - sigmoidf_(-x - 4.0f);
}

// ---------------------------------------------------------------------------
// ragged scatter-add: sums[t, syn_map[ids[t], j]] += syn_mask[ids[t], j]
// ---------------------------------------------------------------------------
__global__ void k_scatter_sums(const int* __restrict__ ids,
                               const int* __restrict__ syn_map,
                               const int* __restrict__ syn_mask,
                               float* __restrict__ sums) {
    int t = blockIdx.x * blockDim.x + threadIdx.x;
    if (t >= T_CTX) return;
    int id = ids[t];
    float* row = sums + (size_t)t * S_DIM;
    const int* mp = syn_map  + (size_t)id * MAXSYN;
    const int* mk = syn_mask + (size_t)id * MAXSYN;
#pragma unroll
    for (int j = 0; j < MAXSYN; ++j) {
        int s = mp[j];
        row[s] += (float)mk[j];    // duplicates handled serially in-thread
    }
}

// ---------------------------------------------------------------------------
// x[:, :512] += emb_v1[ids] ;  x[:, 512:768] = padding
// ---------------------------------------------------------------------------
__global__ void k_build_x(const int* __restrict__ ids,
                          const float* __restrict__ emb_v1,
                          const float* __restrict__ padding,
                          float* __restrict__ x) {
    int i = blockIdx.x * blockDim.x + threadIdx.x;     // exactly T*KX threads
    int t = i / KX;
    int k = i - t * KX;
    if (k < D_DIM) {
        x[i] += emb_v1[(size_t)ids[t] * D_DIM + k];
    } else {
        x[i] = padding[(size_t)t * A_DIM + (k - D_DIM)];
    }
}

// ---------------------------------------------------------------------------
// C[M,N](ldc) = A[M,K] @ B[K,N] + bias[N]
// block tile: 128(M) x 64(N); 8 waves, each 16x64.
// B K-chunks (32 x 64) staged in LDS, K-pair interleaved:
//   element (k, n) lives at Bs[k>>1][2*n + (k&1)]  ->  a lane's fragment
//   (k = kk+2*lk, kk+2*lk+1 ; n = j*16+lm) is one contiguous ds_load_b64.
// requires: M % 128 == 0, N % 64 == 0, K % 32 == 0
// ---------------------------------------------------------------------------
#define BN_PITCH 130   // 64*2 + 2 pad floats per k-pair row (even -> 8B align)
__global__ void k_gemm_bn(const float* __restrict__ A,
                          const float* __restrict__ B,
                          const float* __restrict__ bias,
                          float* __restrict__ C,
                          int M, int N, int K, int ldc) {
    __shared__ float Bs[16 * BN_PITCH];
    int tid  = threadIdx.x;
    int lane = tid & 31;
    int wid  = tid >> 5;                 // 0..7
    int lm   = lane & 15;
    int lk   = lane >> 4;                // K-half select (fp32 16x4 A layout)

    int nblk = N >> 6;
    int bm = blockIdx.x / nblk;
    int bn = blockIdx.x - bm * nblk;
    int m0 = (bm << 7) + (wid << 4);
    int n0 = bn << 6;

    const float* Ap = A + (size_t)(m0 + lm) * K + (lk << 1);

    // cooperative fill: thread owns K-pair p (0..15) x 4 columns
    int p  = tid >> 4;                   // 0..15
    int c0 = (tid & 15) << 2;            // 0..60 step 4
    const float* Bg = B + n0 + c0;

    v8f acc[4] = {};
    for (int kc = 0; kc < K; kc += 32) {
        __syncthreads();
        const float* r0 = Bg + (size_t)(kc + (p << 1)) * N;   // k = kc+2p
        v4f lo = *(const v4f*)(r0);                           // row k
        v4f hi = *(const v4f*)(r0 + N);                       // row k+1
        float* dst = &Bs[p * BN_PITCH + (c0 << 1)];
#pragma unroll
        for (int i = 0; i < 4; ++i) {
            v2f pr;
            pr[0] = lo[i];
            pr[1] = hi[i];
            *(v2f*)(dst + (i << 1)) = pr;                     // ds_store_b64
        }
        __syncthreads();
#pragma unroll
        for (int kk = 0; kk < 32; kk += 4) {
            v2f a = *(const v2f*)(Ap + kc + kk);
            const float* bl = &Bs[((kk >> 1) + lk) * BN_PITCH + (lm << 1)];
#pragma unroll
            for (int j = 0; j < 4; ++j) {
                v2f b = *(const v2f*)(bl + (j << 5));         // ds_load_b64
                acc[j] = WMMA_F32(a, b, acc[j]);
            }
        }
    }
#pragma unroll
    for (int j = 0; j < 4; ++j) {
        int n = n0 + (j << 4) + lm;
        float bz = bias[n];
#pragma unroll
        for (int rr = 0; rr < 8; ++rr) {
            int m = m0 + rr + (lk << 3);
            C[(size_t)m * ldc + n] = acc[j][rr] + bz;
        }
    }
}

// ---------------------------------------------------------------------------
// C[M,N] = A[M,K] @ BT[N,K]^T      (logits = x @ W_rev.T)
// one wave computes 32(M) x 64(N); double-buffered fragment loads so the
// next K-step's loads overlap the current step's 8 WMMAs.
// requires: M % 32 == 0, K % 4 == 0; N edge guarded (V = 50257)
// ---------------------------------------------------------------------------
__global__ void k_gemm_bt(const float* __restrict__ A,
                          const float* __restrict__ BT,
                          float* __restrict__ C,
                          int M, int N, int K) {
    int lane = threadIdx.x & 31;
    int wave = blockIdx.x * (blockDim.x >> 5) + (threadIdx.x >> 5);
    int lm = lane & 15;
    int lk = lane >> 4;
    int ntiles = (N + 63) >> 6;
    int wm = wave / ntiles;
    int wn = wave - wm * ntiles;
    if (wm * 32 >= M) return;
    int m0 = wm << 5, n0 = wn << 6;

    const float* Ap0 = A + (size_t)(m0 + lm) * K + (lk << 1);
    const float* Ap1 = Ap0 + (size_t)16 * K;
    const float* Bp0, *Bp1, *Bp2, *Bp3;
    {
        int n;
        n = n0 +  0 + lm; if (n > N - 1) n = N - 1; Bp0 = BT + (size_t)n * K + (lk << 1);
        n = n0 + 16 + lm; if (n > N - 1) n = N - 1; Bp1 = BT + (size_t)n * K + (lk << 1);
        n = n0 + 32 + lm; if (n > N - 1) n = N - 1; Bp2 = BT + (size_t)n * K + (lk << 1);
        n = n0 + 48 + lm; if (n > N - 1) n = N - 1; Bp3 = BT + (size_t)n * K + (lk << 1);
    }

    v8f acc[8] = {};
    // prologue: first fragments
    v2f a0 = *(const v2f*)Ap0;
    v2f a1 = *(const v2f*)Ap1;
    v2f b0 = *(const v2f*)Bp0;
    v2f b1 = *(const v2f*)Bp1;
    v2f b2 = *(const v2f*)Bp2;
    v2f b3 = *(const v2f*)Bp3;

    for (int k0 = 4; k0 < K; k0 += 4) {
        // issue next-step loads first: they overlap the 8 WMMAs below
        v2f na0 = *(const v2f*)(Ap0 + k0);
        v2f na1 = *(const v2f*)(Ap1 + k0);
        v2f nb0 = *(const v2f*)(Bp0 + k0);
        v2f nb1 = *(const v2f*)(Bp1 + k0);
        v2f nb2 = *(const v2f*)(Bp2 + k0);
        v2f nb3 = *(const v2f*)(Bp3 + k0);
        __builtin_prefetch(Bp0 + k0 + 60, 0, 0);       // global_prefetch_b8

        acc[0] = WMMA_F32(a0, b0, acc[0]);
        acc[1] = WMMA_F32(a0, b1, acc[1]);
        acc[2] = WMMA_F32(a0, b2, acc[2]);
        acc[3] = WMMA_F32(a0, b3, acc[3]);
        acc[4] = WMMA_F32(a1, b0, acc[4]);
        acc[5] = WMMA_F32(a1, b1, acc[5]);
        acc[6] = WMMA_F32(a1, b2, acc[6]);
        acc[7] = WMMA_F32(a1, b3, acc[7]);

        a0 = na0; a1 = na1;
        b0 = nb0; b1 = nb1; b2 = nb2; b3 = nb3;
    }
    // epilogue: last K-step
    acc[0] = WMMA_F32(a0, b0, acc[0]);
    acc[1] = WMMA_F32(a0, b1, acc[1]);
    acc[2] = WMMA_F32(a0, b2, acc[2]);
    acc[3] = WMMA_F32(a0, b3, acc[3]);
    acc[4] = WMMA_F32(a1, b0, acc[4]);
    acc[5] = WMMA_F32(a1, b1, acc[5]);
    acc[6] = WMMA_F32(a1, b2, acc[6]);
    acc[7] = WMMA_F32(a1, b3, acc[7]);

#pragma unroll
    for (int j = 0; j < 4; ++j) {
        int n = n0 + (j << 4) + lm;
        if (n < N) {
#pragma unroll
            for (int rr = 0; rr < 8; ++rr) {
                int m = m0 + rr + (lk << 3);
                C[(size_t)m * (size_t)N + n]        = acc[j][rr];
                C[(size_t)(m + 16) * (size_t)N + n] = acc[4 + j][rr];
            }
        }
    }
}

// ---------------------------------------------------------------------------
extern "C" void kernel_launch(void* const* d_in, const int* in_sizes, int n_in,
                              void* d_out, int out_size, void* d_ws, size_t ws_size,
                              hipStream_t stream) {
    const int*   ids      = (const int*)d_in[0];
    const int*   syn_map  = (const int*)d_in[1];
    const int*   syn_mask = (const int*)d_in[2];
    const float* emb_v1   = (const float*)d_in[3];
    const float* emb_v2   = (const float*)d_in[4];
    const float* W_out    = (const float*)d_in[5];
    const float* b_out    = (const float*)d_in[6];
    const float* W_sum    = (const float*)d_in[7];
    const float* b_sum    = (const float*)d_in[8];
    const float* W_rev    = (const float*)d_in[9];
    const float* padding  = (const float*)d_in[10];

    float* out     = (float*)d_out;
    float* logits  = out;                                  // [T, V]
    float* g       = logits + (size_t)T_CTX * V_VOCAB;     // [T, D]
    float* sums    = g + (size_t)T_CTX * D_DIM;            // [T, S]
    float* exp_sum = sums + (size_t)T_CTX * S_DIM;         // [T, S]
    float* x       = (float*)d_ws;                         // [T, 768] scratch

    // 1. g = three_stage(emb_v2[ids])
    k_gather_g<<<(T_CTX * D_DIM) / 256, 256, 0, stream>>>(ids, emb_v2, g);

    // 2. sums: zero then ragged scatter-add
    hipMemsetAsync(sums, 0, (size_t)T_CTX * S_DIM * sizeof(float), stream);
    k_scatter_sums<<<T_CTX / 256, 256, 0, stream>>>(ids, syn_map, syn_mask, sums);

    // 3. exp_sum = g @ W_sum + b_sum          [4096 x 8192], K=512
    {
        int blocks = (T_CTX / 128) * (S_DIM / 64);         // 32*128 = 4096
        k_gemm_bn<<<blocks, 256, 0, stream>>>(g, W_sum, b_sum, exp_sum,
                                              T_CTX, S_DIM, D_DIM, S_DIM);
    }

    // 4. x[:, :512] = g @ W_out + b_out       [4096 x 512], K=512, ldc=768
    {
        int blocks = (T_CTX / 128) * (D_DIM / 64);         // 32*8 = 256
        k_gemm_bn<<<blocks, 256, 0, stream>>>(g, W_out, b_out, x,
                                              T_CTX, D_DIM, D_DIM, KX);
    }

    // 5. x[:, :512] += emb_v1[ids];  x[:, 512:768] = padding
    k_build_x<<<(T_CTX * KX) / 256, 256, 0, stream>>>(ids, emb_v1, padding, x);

    // 6. logits = x @ W_rev^T                 [4096 x 50257], K=768
    {
        int ntiles = (V_VOCAB + 63) / 64;                  // 786
        int waves  = (T_CTX / 32) * ntiles;                // 128*786 = 100608
        int blocks = waves / 8;                            // 12576
        k_gemm_bt<<<blocks, 256, 0, stream>>>(x, W_rev, logits,
                                              T_CTX, V_VOCAB, KX);
    }
}